// TUNED_42159398978046
// MI455X (gfx1250) — compile-verified
//
#include <hip/hip_runtime.h>
#include <hip/hip_bf16.h>

// ---------------------------------------------------------------------------
// MI455X (gfx1250) implementation.
// Heavy ops via V_WMMA_F32_16X16X4_F32 (wave32 fp32 matrix path).
// A (201 MB) streamed with non-temporal loads (keeps L2 for the reused B
// operands); each block owns a 64-row x 512-col output strip so A is read
// exactly once per pass and B re-reads (N/64 per view) hit L2.
// ---------------------------------------------------------------------------

constexpr int V = 3;
constexpr int N = 4096;
constexpr int D = 512;
constexpr int C = 10;
constexpr float PSI = 0.7f;

typedef __attribute__((ext_vector_type(2))) float v2f;
typedef __attribute__((ext_vector_type(4))) float v4f;
typedef __attribute__((ext_vector_type(8))) float v8f;

__device__ __forceinline__ float softplusf(float x) {
    return (x > 20.0f) ? x : log1pf(expf(x));
}

// ---------------------------------------------------------------------------
// dinv[row] = 1/sqrt(1 + sum_j A[row, j])   for row in [0, V*N)
// A streamed non-temporally (201 MB, single use).
// ---------------------------------------------------------------------------
__global__ __launch_bounds__(256) void k_dinv(const float* __restrict__ A,
                                              float* __restrict__ dinv) {
    __shared__ float red[256];
    const long row = blockIdx.x;                 // 0 .. V*N-1 ([V,N,N] flat)
    const v4f* a = (const v4f*)(A + row * (long)N);
    float s = 0.0f;
    for (int j = threadIdx.x; j < N / 4; j += 256) {
        const v4f t = __builtin_nontemporal_load(a + j);
        s += t.x + t.y + t.z + t.w;
    }
    red[threadIdx.x] = s;
    __syncthreads();
    for (int off = 128; off > 0; off >>= 1) {
        if (threadIdx.x < off) red[threadIdx.x] += red[threadIdx.x + off];
        __syncthreads();
    }
    if (threadIdx.x == 0) dinv[row] = rsqrtf(red[0] + 1.0f);
}

// ---------------------------------------------------------------------------
// Y[idx] = X[idx] * dinv[row]   (row-scale a [V,N,D] tensor), float4 wide
// ---------------------------------------------------------------------------
__global__ __launch_bounds__(256) void k_rowscale(const float* __restrict__ X,
                                                  const float* __restrict__ dinv,
                                                  float* __restrict__ Y) {
    const long i4 = (long)blockIdx.x * 256 + threadIdx.x;    // < V*N*D/4
    const long row = i4 >> 7;                                // (i4*4)/D
    const v4f x = *((const v4f*)X + i4);
    *((v4f*)Y + i4) = x * dinv[row];
}

// ---------------------------------------------------------------------------
// WMMA fp32 GEMM: acc = Av[M x K] @ Bv[K x 512]   (per-view, blockIdx.y = v)
// Block = 256 thr (8 waves); block owns 64 rows x all 512 cols; wave w owns
// cols [64w, 64w+64) over 4x4 = 16 WMMA accum tiles (128 VGPRs).
// A tile (64 x 64) staged in LDS via non-temporal b128 loads.
//
// mode 0 (GCN):   out = dinv[i]*acc + dinv[i]^2 * diagM[i,:]
// mode 1 (dense): h = acc + bias[col]; out = h; out2 = dinv[i]*h
// ---------------------------------------------------------------------------
#define MT 64
#define KTILE 64
#define LDSP 68   // row stride: 16B-aligned rows, conflict-free column reads

__global__ __launch_bounds__(256) void gemm64_wmma_f32(
    const float* __restrict__ A, long aViewStride, int lda, int K,
    const float* __restrict__ B, long bViewStride,
    const float* __restrict__ dinv,
    const float* __restrict__ diagM,   // mode 0
    const float* __restrict__ bias,    // mode 1
    float* __restrict__ out,
    float* __restrict__ out2,          // mode 1
    int mode) {
    __shared__ float As[MT][LDSP];               // 64 x 68 x 4B = 17.4 KB

    const int v    = blockIdx.y;
    const int m0   = blockIdx.x * MT;
    const int wave = threadIdx.x >> 5;
    const int lane = threadIdx.x & 31;
    const int n0   = wave * 64;

    const float* Av = A + (long)v * aViewStride;
    const float* Bv = B + (long)v * bViewStride;

    v8f acc[4][4] = {};                          // 16 x (16x16 f32), 128 VGPRs

    const int rla = lane >> 4;                   // lane half -> K-pair / M-half
    const int l16 = lane & 15;

    for (int k0 = 0; k0 < K; k0 += KTILE) {
        // stage A tile: 64 rows x 64 cols, b128 non-temporal, coalesced
        for (int i = threadIdx.x; i < MT * (KTILE / 4); i += 256) {
            const int r  = i >> 4;
            const int c4 = (i & 15) * 4;
            const v4f t = __builtin_nontemporal_load(
                (const v4f*)&Av[(long)(m0 + r) * lda + (k0 + c4)]);
            *(v4f*)&As[r][c4] = t;
        }
        __syncthreads();

        for (int kk = 0; kk < KTILE; kk += 4) {
            // A frags 16x4: VGPR0 = K {0|2}, VGPR1 = K {1|3} by lane half
            v2f a[4];
            #pragma unroll
            for (int mt = 0; mt < 4; ++mt) {
                a[mt].x = As[mt * 16 + l16][kk + 2 * rla];
                a[mt].y = As[mt * 16 + l16][kk + 2 * rla + 1];
            }
            const long brow = (long)(k0 + kk + 2 * rla);
            #pragma unroll
            for (int st = 0; st < 4; ++st) {
                const int bcol = n0 + st * 16 + l16;
                v2f b;
                b.x = Bv[brow * 512 + bcol];
                b.y = Bv[(brow + 1) * 512 + bcol];
                #pragma unroll
                for (int mt = 0; mt < 4; ++mt) {
                    acc[mt][st] = __builtin_amdgcn_wmma_f32_16x16x4_f32(
                        false, a[mt], false, b, (short)0, acc[mt][st],
                        false, false);
                }
            }
        }
        __syncthreads();
    }

    // epilogue: C/D layout — VGPR r, lanes 0-15 -> row r, lanes 16-31 -> r+8
    const long outView = (long)v * (long)N * 512;
    #pragma unroll
    for (int mt = 0; mt < 4; ++mt) {
        #pragma unroll
        for (int st = 0; st < 4; ++st) {
            #pragma unroll
            for (int r = 0; r < 8; ++r) {
                const int row = m0 + mt * 16 + r + 8 * rla;
                const int col = n0 + st * 16 + l16;
                const long o = outView + (long)row * 512 + col;
                const float a = acc[mt][st][r];
                if (mode == 0) {
                    const float di = dinv[v * N + row];
                    out[o] = di * a + di * di * diagM[o];
                } else {
                    const float h = a + bias[v * 512 + col];
                    out[o] = h;
                    out2[o] = dinv[v * N + row] * h;
                }
            }
        }
    }
}

// ---------------------------------------------------------------------------
// E[v,n,:] = consensus + softplus(X@Wm + bm) + softplus(h3@W2 + b2)
// One wave32 per (v,n); C=10 accumulators reduced with __shfl_down.
// ---------------------------------------------------------------------------
__global__ __launch_bounds__(256) void k_evidence(
    const float* __restrict__ X, const float* __restrict__ H3,
    const float* __restrict__ Wm, const float* __restrict__ bm,
    const float* __restrict__ W2, const float* __restrict__ b2,
    const float* __restrict__ cons, float* __restrict__ E) {
    const int wave = threadIdx.x >> 5;
    const int lane = threadIdx.x & 31;
    const int g = blockIdx.x * 8 + wave;          // wave id -> (v, n)
    const int v = g >> 12;                        // N = 4096
    const int n = g & (N - 1);

    const float* x  = X  + ((long)v * N + n) * D;
    const float* h  = H3 + ((long)v * N + n) * D;
    const float* wm = Wm + (long)v * D * C;
    const float* w2 = W2 + (long)v * D * C;

    float am[C] = {}, ag[C] = {};
    for (int d = lane; d < D; d += 32) {
        const float xv = x[d], hv = h[d];
        #pragma unroll
        for (int c = 0; c < C; ++c) {
            am[c] += xv * wm[d * C + c];
            ag[c] += hv * w2[d * C + c];
        }
    }
    #pragma unroll
    for (int c = 0; c < C; ++c) {
        for (int off = 16; off > 0; off >>= 1) {
            am[c] += __shfl_down(am[c], off, 32);
            ag[c] += __shfl_down(ag[c], off, 32);
        }
    }
    if (lane == 0) {
        const long base = ((long)v * N + n) * C;
        #pragma unroll
        for (int c = 0; c < C; ++c) {
            const float m = softplusf(am[c] + bm[v * C + c]);
            const float gg = softplusf(ag[c] + b2[v * C + c]);
            E[base + c] = cons[base + c] + m + gg;
        }
    }
}

// ---------------------------------------------------------------------------
// One block: S[i,j] = mean_n cos(E_i[n], E_j[n]) for pairs (0,1),(0,2),(1,2);
// then thread 0 does threshold + per-row masked softmax + (colsum+rowsum)/2V.
// Deterministic (fixed-order LDS reduction, no atomics).
// ---------------------------------------------------------------------------
__global__ __launch_bounds__(256) void k_sim_mrf(const float* __restrict__ E,
                                                 float* __restrict__ wv) {
    __shared__ float r01[256], r02[256], r12[256];
    float s01 = 0.f, s02 = 0.f, s12 = 0.f;
    for (int n = threadIdx.x; n < N; n += 256) {
        float e0[C], e1[C], e2[C];
        #pragma unroll
        for (int c = 0; c < C; ++c) {
            e0[c] = E[(long)n * C + c];
            e1[c] = E[((long)N + n) * C + c];
            e2[c] = E[((long)2 * N + n) * C + c];
        }
        float q0 = 0.f, q1 = 0.f, q2 = 0.f, d01 = 0.f, d02 = 0.f, d12 = 0.f;
        #pragma unroll
        for (int c = 0; c < C; ++c) {
            q0 += e0[c] * e0[c]; q1 += e1[c] * e1[c]; q2 += e2[c] * e2[c];
            d01 += e0[c] * e1[c]; d02 += e0[c] * e2[c]; d12 += e1[c] * e2[c];
        }
        const float n0 = fmaxf(sqrtf(q0), 1e-8f);
        const float n1 = fmaxf(sqrtf(q1), 1e-8f);
        const float n2 = fmaxf(sqrtf(q2), 1e-8f);
        s01 += d01 / (n0 * n1);
        s02 += d02 / (n0 * n2);
        s12 += d12 / (n1 * n2);
    }
    r01[threadIdx.x] = s01; r02[threadIdx.x] = s02; r12[threadIdx.x] = s12;
    __syncthreads();
    for (int off = 128; off > 0; off >>= 1) {
        if (threadIdx.x < off) {
            r01[threadIdx.x] += r01[threadIdx.x + off];
            r02[threadIdx.x] += r02[threadIdx.x + off];
            r12[threadIdx.x] += r12[threadIdx.x + off];
        }
        __syncthreads();
    }
    if (threadIdx.x == 0) {
        const float S01 = r01[0] / N, S02 = r02[0] / N, S12 = r12[0] / N;
        float S[3][3] = {{0.f, S01, S02}, {0.f, 0.f, S12}, {0.f, 0.f, 0.f}};
        const bool mask[3][3] = {{false, true, true},
                                 {false, false, true},
                                 {false, false, false}};
        const float thr = PSI * fmaxf(S01, fmaxf(S02, S12));
        float w[3][3], p[3][3];
        for (int i = 0; i < 3; ++i)
            for (int j = 0; j < 3; ++j)
                w[i][j] = (mask[i][j] && S[i][j] > thr) ? S[i][j] : 0.f;
        for (int i = 0; i < 3; ++i) {
            float m = -1e30f;
            for (int j = 0; j < 3; ++j) {
                const float l = mask[i][j] ? w[i][j] : -1e30f;
                m = fmaxf(m, l);
            }
            float e[3], sum = 0.f;
            for (int j = 0; j < 3; ++j) {
                const float l = mask[i][j] ? w[i][j] : -1e30f;
                e[j] = mask[i][j] ? expf(l - m) : 0.f;
                sum += e[j];
            }
            const float den = fmaxf(sum, 1e-30f);
            for (int j = 0; j < 3; ++j) p[i][j] = e[j] / den;
        }
        for (int x = 0; x < 3; ++x) {
            float colsum = 0.f, rowsum = 0.f;
            for (int i = 0; i < 3; ++i) { colsum += p[i][x]; rowsum += p[x][i]; }
            wv[x] = (colsum + rowsum) / (2.0f * V);
        }
    }
}

// ---------------------------------------------------------------------------
// agg[n,c] = sum_v wv[v] * E[v,n,c]
// ---------------------------------------------------------------------------
__global__ __launch_bounds__(256) void k_agg(const float* __restrict__ E,
                                             const float* __restrict__ wv,
                                             float* __restrict__ agg) {
    const int idx = blockIdx.x * 256 + threadIdx.x;   // < N*C
    float s = 0.f;
    #pragma unroll
    for (int v = 0; v < V; ++v) s += wv[v] * E[(long)v * N * C + idx];
    agg[idx] = s;
}

// ---------------------------------------------------------------------------
extern "C" void kernel_launch(void* const* d_in, const int* in_sizes, int n_in,
                              void* d_out, int out_size, void* d_ws, size_t ws_size,
                              hipStream_t stream) {
    (void)in_sizes; (void)n_in; (void)out_size; (void)ws_size;
    const float* X     = (const float*)d_in[0];
    const float* A     = (const float*)d_in[1];
    const float* cons  = (const float*)d_in[2];
    const float* W_mlp = (const float*)d_in[3];
    const float* b_mlp = (const float*)d_in[4];
    const float* W1    = (const float*)d_in[5];
    const float* b1    = (const float*)d_in[6];
    const float* W2    = (const float*)d_in[7];
    const float* b2    = (const float*)d_in[8];

    float* out = (float*)d_out;
    float* E   = out;                             // [V,N,C]
    float* agg = out + (long)V * N * C;           // [N,C]

    const long BUF = (long)V * N * D;             // 6,291,456 floats
    float* ws   = (float*)d_ws;
    float* dinv = ws;                             // V*N
    float* wv   = ws + (long)V * N;               // 16 (3 used)
    float* B0   = wv + 16;                        // Y1, later Y2
    float* B1   = B0 + BUF;                       // h1, later h3
    float* B2   = B1 + BUF;                       // h2

    // 1) degree normalization (NT stream over A)
    k_dinv<<<V * N, 256, 0, stream>>>(A, dinv);
    // 2) Y1 = dinv (*) X
    k_rowscale<<<(int)(BUF / 4 / 256), 256, 0, stream>>>(X, dinv, B0);

    const dim3 g(N / MT, V);
    // 3) h1 = dinv*(A@Y1) + dinv^2*X
    gemm64_wmma_f32<<<g, 256, 0, stream>>>(A, (long)N * N, N, N,
                                           B0, (long)N * D,
                                           dinv, X, nullptr, B1, nullptr, 0);
    // 4) h2 = h1@W1 + b1 ; Y2 = dinv (*) h2
    gemm64_wmma_f32<<<g, 256, 0, stream>>>(B1, (long)N * D, D, D,
                                           W1, (long)D * D,
                                           dinv, nullptr, b1, B2, B0, 1);
    // 5) h3 = dinv*(A@Y2) + dinv^2*h2
    gemm64_wmma_f32<<<g, 256, 0, stream>>>(A, (long)N * N, N, N,
                                           B0, (long)N * D,
                                           dinv, B2, nullptr, B1, nullptr, 0);
    // 6) E = consensus + softplus(X@Wm+bm) + softplus(h3@W2+b2)
    k_evidence<<<(V * N) / 8, 256, 0, stream>>>(X, B1, W_mlp, b_mlp, W2, b2,
                                                cons, E);
    // 7) pairwise cosine sims + MRF softmax -> per-view weights
    k_sim_mrf<<<1, 256, 0, stream>>>(E, wv);
    // 8) agg
    k_agg<<<(N * C) / 256, 256, 0, stream>>>(E, wv, agg);
}